// WeightsEllipse_20701742367195
// MI455X (gfx1250) — compile-verified
//
#include <hip/hip_runtime.h>

// WeightsEllipse on gfx1250:
//   out[oi][pix] = exp( - ( Coeff[oi][0:6] . Basis[0:6][pix] ) ^ a[oi] )
// Polynomial contraction (K=6) done with two V_WMMA_F32_16X16X4_F32 steps,
// epilogue with hardware v_log_f32 / v_exp_f32. Store-bandwidth bound (64MB).

typedef float v2f __attribute__((ext_vector_type(2)));
typedef float v8f __attribute__((ext_vector_type(8)));

#define NOI   4096   // O*I columns (oi pairs)
#define NPIX  4096   // GH*GW pixels
#define PAR_S 7      // floats per (o,i): mx,my,s00,s01,s10,s11,a

__global__ __launch_bounds__(256)
void weights_ellipse_wmma(const float* __restrict__ param,
                          float* __restrict__ out)
{
    const int lane        = threadIdx.x & 31;
    const int wave_in_blk = threadIdx.x >> 5;
    const int wave_id     = blockIdx.x * 8 + wave_in_blk;  // 8192 waves total
    const int oi_tile     = wave_id >> 5;                  // 0..255 (16 oi rows each)
    const int chunk       = wave_id & 31;                  // 0..31  (128 pixels each)
    const int oi_base     = oi_tile * 16;
    const int hsel        = lane >> 4;                     // 0: lanes 0-15, 1: lanes 16-31
    const int m           = lane & 15;

    // ---- per-wave A-fragment setup: polynomial coefficients for 16 oi rows ----
    // q(u,v) = c0*u^2 + c1*u*v + c2*v^2 + c3*u + c4*v + c5
    const float* pr = param + (size_t)(oi_base + m) * PAR_S;
    const float mx  = pr[0], my = pr[1];
    const float s00 = pr[2], s01 = pr[3], s10 = pr[4], s11 = pr[5];
    const float sxy = s01 + s10;                // reference uses full (non-sym) form
    const float c0  = s00;
    const float c1  = sxy;
    const float c2  = s11;
    const float c3  = -(2.0f * s00 * mx + sxy * my);
    const float c4  = -(sxy * mx + 2.0f * s11 * my);
    const float c5  = s00*mx*mx + sxy*mx*my + s11*my*my;

    // A 16x4 f32 layout (ISA 7.12.2): lanes 0-15 hold K=0,1; lanes 16-31 hold K=2,3.
    // Step 1 basis K = {u^2, u*v, v^2, u}; step 2 basis K = {v, 1, 0, 0}.
    v2f A1, A2;
    A1.x = hsel ? c2 : c0;
    A1.y = hsel ? c3 : c1;
    A2.x = hsel ? 0.0f : c4;
    A2.y = hsel ? 0.0f : c5;

    // Exponent a for the 8 D-matrix rows this lane holds (M = j + 8*hsel).
    float arow[8];
#pragma unroll
    for (int j = 0; j < 8; ++j)
        arow[j] = param[(size_t)(oi_base + j + hsel * 8) * PAR_S + 6];

    const float NEG_LOG2E = -1.4426950408889634f;

#pragma unroll 1
    for (int t = 0; t < 8; ++t) {
        const int   pbase = (chunk * 8 + t) * 16;
        const int   p     = pbase + m;            // this lane's pixel column
        const float u     = (float)(p >> 6);      // row index (p // 64)
        const float v     = (float)(p & 63);      // col index (p %  64)

        // B 4x16 f32 layout: VGPR j, lane l -> K = j + 2*(l>=16), N = l%16.
        v2f B1, B2;
        B1.x = hsel ? (v * v) : (u * u);   // K=0: u^2 | K=2: v^2
        B1.y = hsel ? u       : (u * v);   // K=1: u*v | K=3: u
        B2.x = hsel ? 0.0f    : v;         // K=0: v   | K=2: 0
        B2.y = hsel ? 0.0f    : 1.0f;      // K=1: 1   | K=3: 0

        v8f acc = {};
        acc = __builtin_amdgcn_wmma_f32_16x16x4_f32(false, A1, false, B1,
                                                    (short)0, acc, false, false);
        acc = __builtin_amdgcn_wmma_f32_16x16x4_f32(false, A2, false, B2,
                                                    (short)0, acc, false, false);

        // Epilogue: w = exp(-q^a) = exp2( -log2e * exp2( a * log2(q) ) )
        // D layout: VGPR j, lane l -> row M = j + 8*(l>=16), col N = l%16.
#pragma unroll
        for (int j = 0; j < 8; ++j) {
            float q  = acc[j];
            q        = fmaxf(q, 0.0f);  // guard rounding; q=0 -> w=1 via -inf path
            float qa = __builtin_amdgcn_exp2f(arow[j] * __builtin_amdgcn_logf(q));
            float w  = __builtin_amdgcn_exp2f(qa * NEG_LOG2E);
            out[(size_t)(oi_base + j + hsel * 8) * NPIX + pbase + m] = w;
        }
    }
}

extern "C" void kernel_launch(void* const* d_in, const int* in_sizes, int n_in,
                              void* d_out, int out_size, void* d_ws, size_t ws_size,
                              hipStream_t stream) {
    const float* param = (const float*)d_in[0];
    float*       out   = (float*)d_out;

    // 8192 waves = 1024 blocks x 8 waves (256 threads, wave32).
    dim3 grid(1024), block(256);
    hipLaunchKernelGGL(weights_ellipse_wmma, grid, block, 0, stream, param, out);
}